// DiffeomorphicTransform_23106924052584
// MI455X (gfx1250) — compile-verified
//
#include <hip/hip_runtime.h>

// DiffeomorphicTransform: scaling-and-squaring of a velocity field.
// B=1, C=3, D=128, H=160, W=128. flow layout: [C, D*H*W] (channel-major).
// sample_grid layout: [D*H*W, 3] (xyz interleaved).
//
// Bandwidth-bound gather op (no matrix content -> no WMMA). MI455X plan:
//  - whole working set (velocity 31.5MB + grid 31.5MB + 2x flow 63MB = 126MB)
//    is resident in the 192MB L2 after first touch; default RT hints keep it there.
//  - x-neighbor corner pairs fetched as single b64 gathers (12 instead of 24
//    b32 gathers per voxel) -> halves L2 gather request count.
//  - gfx1250 global_prefetch_b8 on the sequential streams.

#define D_ 128
#define H_ 160
#define W_ 128
constexpr int N_ = D_ * H_ * W_;          // 2,621,440 voxels
constexpr int TIME_STEP_ = 7;

struct F2 { float lo, hi; };              // natural align = 4 -> legal 4B-aligned b64

__global__ __launch_bounds__(256) void dvf_step_kernel(
    const float* __restrict__ src,    // flow in  [3, N_] (raw; multiply by inScale)
    const float* __restrict__ grid,   // sample_grid [N_, 3]
    const float* __restrict__ rf_p,   // range_flow scalar (1 element)
    float* __restrict__ dst,          // flow out [3, N_]
    float inScale)
{
    const int n = blockIdx.x * blockDim.x + threadIdx.x;
    if (n >= N_) return;

    const float rf = rf_p[0];

    // gfx1250 prefetch (global_prefetch_b8) on the sequential streams.
    if (n + 16384 < N_) {
        __builtin_prefetch(grid + 3 * (n + 16384), 0, 1);
        __builtin_prefetch(src  + (n + 16384),     0, 1);
    }

    // Center flow vector (scaled)
    const float f0 = inScale * src[n];
    const float f1 = inScale * src[n + N_];
    const float f2 = inScale * src[n + 2 * N_];

    // Sampling position in normalized coords
    const float gx = grid[3 * n + 0] + f0 * rf;
    const float gy = grid[3 * n + 1] + f1 * rf;
    const float gz = grid[3 * n + 2] + f2 * rf;

    // Unnormalize (align_corners=True) + border clamp
    float ix = (gx + 1.0f) * (0.5f * (float)(W_ - 1));
    float iy = (gy + 1.0f) * (0.5f * (float)(H_ - 1));
    float iz = (gz + 1.0f) * (0.5f * (float)(D_ - 1));
    ix = fminf(fmaxf(ix, 0.0f), (float)(W_ - 1));
    iy = fminf(fmaxf(iy, 0.0f), (float)(H_ - 1));
    iz = fminf(fmaxf(iz, 0.0f), (float)(D_ - 1));

    const float x0f = floorf(ix);
    const float y0f = floorf(iy);
    const float z0f = floorf(iz);
    const float wx = ix - x0f;
    const float wy = iy - y0f;
    const float wz = iz - z0f;

    const int x0 = (int)x0f;
    const int y0 = (int)y0f;
    const int z0 = (int)z0f;
    const int y1 = (y0 + 1 < H_) ? y0 + 1 : H_ - 1;
    const int z1 = (z0 + 1 < D_) ? z0 + 1 : D_ - 1;

    // x-pair base: load (s[xb], s[xb+1]) as one b64. When x0 == W-1 the sample
    // clamped onto the border (wx == 0); select v0 = hi so the result is exact.
    const bool xEdge = (x0 == W_ - 1);
    const int  xb    = xEdge ? (W_ - 2) : x0;

    // Row base offsets for the 4 (z,y) corner rows
    const int r00 = (z0 * H_ + y0) * W_ + xb;
    const int r01 = (z0 * H_ + y1) * W_ + xb;
    const int r10 = (z1 * H_ + y0) * W_ + xb;
    const int r11 = (z1 * H_ + y1) * W_ + xb;

    const float fc[3] = {f0, f1, f2};

#pragma unroll
    for (int c = 0; c < 3; ++c) {
        const float* __restrict__ s = src + c * N_;
        const F2 p00 = *reinterpret_cast<const F2*>(s + r00);
        const F2 p01 = *reinterpret_cast<const F2*>(s + r01);
        const F2 p10 = *reinterpret_cast<const F2*>(s + r10);
        const F2 p11 = *reinterpret_cast<const F2*>(s + r11);

        const float v00 = xEdge ? p00.hi : p00.lo;   // (wx==0 on edge -> exact)
        const float v01 = xEdge ? p01.hi : p01.lo;
        const float v10 = xEdge ? p10.hi : p10.lo;
        const float v11 = xEdge ? p11.hi : p11.lo;

        const float a0 = v00 + wx * (p00.hi - v00);
        const float a1 = v01 + wx * (p01.hi - v01);
        const float b0 = v10 + wx * (p10.hi - v10);
        const float b1 = v11 + wx * (p11.hi - v11);
        const float c0 = a0 + wy * (a1 - a0);
        const float c1 = b0 + wy * (b1 - b0);
        const float sv = c0 + wz * (c1 - c0);

        dst[c * N_ + n] = fc[c] + inScale * sv;
    }
}

extern "C" void kernel_launch(void* const* d_in, const int* in_sizes, int n_in,
                              void* d_out, int out_size, void* d_ws, size_t ws_size,
                              hipStream_t stream) {
    (void)in_sizes; (void)n_in; (void)out_size; (void)ws_size;

    const float* velocity = (const float*)d_in[0];  // [3, N_]
    const float* grid     = (const float*)d_in[1];  // [N_, 3]
    const float* rf       = (const float*)d_in[2];  // scalar

    float* out = (float*)d_out;   // [3, N_] ping buffer (final result lands here)
    float* ws  = (float*)d_ws;    // [3, N_] pong buffer (needs 31.5 MB)

    const dim3 block(256);
    const dim3 gridDim((N_ + 255) / 256);

    // Step 1: reads raw velocity, folds /2^TIME_STEP via inScale, writes d_out.
    const float s0 = 1.0f / (float)(1 << TIME_STEP_);
    dvf_step_kernel<<<gridDim, block, 0, stream>>>(velocity, grid, rf, out, s0);

    // Steps 2..7: ping-pong out <-> ws; with 6 remaining steps the final
    // write (step 7) lands back in d_out.
    for (int it = 1; it < TIME_STEP_; ++it) {
        const float* srcp = (it & 1) ? out : ws;
        float*       dstp = (it & 1) ? ws  : out;
        dvf_step_kernel<<<gridDim, block, 0, stream>>>(srcp, grid, rf, dstp, 1.0f);
    }
}